// RASPScoreModule_395136991936
// MI455X (gfx1250) — compile-verified
//
#include <hip/hip_runtime.h>
#include <hip/hip_bf16.h>

typedef float v2f __attribute__((ext_vector_type(2)));
typedef float v8f __attribute__((ext_vector_type(8)));

#define T_TYPES 85
#define D_BINS  21
#define EXCL    2

__global__ void rasp_zero_kernel(float* out) { out[0] = 0.0f; }

__global__ __launch_bounds__(256) void rasp_score_kernel(
    const float* __restrict__ coords,
    const int*   __restrict__ res_ids,
    const int*   __restrict__ types,
    const float* __restrict__ pot,
    float*       __restrict__ out,
    int n)
{
    const int lane       = threadIdx.x & 31;
    const int waveInBlk  = threadIdx.x >> 5;
    const int wavesPerBlk = blockDim.x >> 5;
    const int gwave      = blockIdx.x * wavesPerBlk + waveInBlk;
    const int totWaves   = gridDim.x * wavesPerBlk;

    const int NT  = (n + 15) >> 4;       // 16x16 tiles per side
    const int col = lane & 15;
    const int hi  = lane >> 4;           // 0: lanes 0-15, 1: lanes 16-31

    float acc = 0.0f;

    for (int t = gwave; t < NT * NT; t += totWaves) {
        const int ti = t / NT;
        const int tj = t - ti * NT;
        if (tj < ti) continue;           // upper-triangular tiles only

        const int i_at = ti * 16 + col;
        const int j_at = tj * 16 + col;
        const int il = (i_at < n) ? i_at : (n - 1);
        const int jl = (j_at < n) ? j_at : (n - 1);

        const float ix = coords[3 * il + 0];
        const float iy = coords[3 * il + 1];
        const float iz = coords[3 * il + 2];
        const float jx = coords[3 * jl + 0];
        const float jy = coords[3 * jl + 1];
        const float jz = coords[3 * jl + 2];
        const int   res_i  = res_ids[il];
        const int   type_i = types[il];
        const int   res_j  = res_ids[jl];
        const int   type_j = types[jl];
        const float nsq_i = ix * ix + iy * iy + iz * iz;
        const float nsq_j = jx * jx + jy * jy + jz * jz;

        // A (16x4 f32): lanes 0-15 hold K=0,1 ; lanes 16-31 hold K=2,3.
        // Row i of A = (x_i, y_i, z_i, 1)
        v2f a;
        a.x = hi ? iz   : ix;
        a.y = hi ? 1.0f : iy;
        // B (4x16 f32): mirrored striping. Col j of B = (-2x_j, -2y_j, -2z_j, |c_j|^2)
        v2f b;
        b.x = hi ? (-2.0f * jz) : (-2.0f * jx);
        b.y = hi ? nsq_j        : (-2.0f * jy);

        v8f c = {};
        // C[m][nn] = |c_j|^2 - 2 c_i . c_j   (add |c_i|^2 per row below)
        c = __builtin_amdgcn_wmma_f32_16x16x4_f32(
                false, a, false, b, (short)0, c, false, false);

        const int j_glob = tj * 16 + col;

        #pragma unroll
        for (int r = 0; r < 8; ++r) {
            const int m     = r + 8 * hi;        // tile row of this accumulator
            const int i_row = ti * 16 + m;       // global i index

            // row-atom metadata lives in lane m (it loaded atom ti*16+m)
            const int   res_im  = __shfl(res_i,  m, 32);
            const int   type_im = __shfl(type_i, m, 32);
            const float nsq_im  = __shfl(nsq_i,  m, 32);

            int sep = res_im - res_j;
            sep = (sep < 0) ? -sep : sep;

            bool valid = (j_glob > i_row) && (j_glob < n) && (i_row < n)
                         && (sep > EXCL) && (type_im != -1) && (type_j != -1);

            const float d2   = c[r] + nsq_im;
            const float dist = sqrtf(fmaxf(d2, 0.0f)) + 1e-8f;
            valid = valid && (dist < (float)(D_BINS - 1));

            if (valid) {
                int k = sep - 1;
                if (k > 5) k = 5;
                const int   d0    = (int)dist;            // dist in [0, 20)
                const float alpha = dist - (float)d0;
                const int base = ((k * T_TYPES + type_im) * T_TYPES + type_j) * D_BINS + d0;
                const float e0 = pot[base];
                const float e1 = pot[base + 1];
                acc += (1.0f - alpha) * e0 + alpha * e1 - 2.7f;
            }
        }
    }

    // wave32 reduction
    #pragma unroll
    for (int off = 16; off > 0; off >>= 1)
        acc += __shfl_down(acc, off, 32);

    __shared__ float s_part[8];
    if (lane == 0) s_part[waveInBlk] = acc;
    __syncthreads();

    if (waveInBlk == 0) {
        float v = (lane < wavesPerBlk) ? s_part[lane] : 0.0f;
        #pragma unroll
        for (int off = 16; off > 0; off >>= 1)
            v += __shfl_down(v, off, 32);
        if (lane == 0) atomicAdd(out, v);
    }
}

extern "C" void kernel_launch(void* const* d_in, const int* in_sizes, int n_in,
                              void* d_out, int out_size, void* d_ws, size_t ws_size,
                              hipStream_t stream) {
    const float* coords  = (const float*)d_in[0];
    const int*   res_ids = (const int*)d_in[1];
    const int*   types   = (const int*)d_in[2];
    const float* pot     = (const float*)d_in[3];
    float*       out     = (float*)d_out;

    const int n = in_sizes[1];           // number of atoms

    rasp_zero_kernel<<<1, 1, 0, stream>>>(out);

    const int threads = 256;             // 8 waves per block
    const int blocks  = 256;             // 2048 waves total
    rasp_score_kernel<<<blocks, threads, 0, stream>>>(
        coords, res_ids, types, pot, out, n);
}